// VMD_block_62612033241298
// MI455X (gfx1250) — compile-verified
//
#include <hip/hip_runtime.h>
#include <math.h>

// ---------------------------------------------------------------------------
// MVMD (K=4, ALPHA=2000, TAU=0, TOL=5e-5, MAX_N=5) on x:(8,2048,64) f32.
// 512 signals of length 2048, mirror-extended to 4096.
// FFT(4096) = radix-16^3, each radix-16 stage done as complex 16x16 matmul
// with the DFT-16 matrix on the WMMA units (v_wmma_f32_16x16x4_f32).
// Inverse FFT = conj -> forward FFT -> Re()/N.
//
// Workspace layout (needs ~64MB + 64B):
//   buf0 : 512*4096 float2  (16MB)
//   buf1 : 512*4096 float2  (16MB)   (also aliased as f_hat_plus, stride 4096)
//   u    : 4*512*2048 float2 (32MB)  (positive-half spectra of the 4 modes)
//   sc   : 16 floats  [0..3]=omega [4..7]=num [8..11]=den [12]=diff [13]=done
// ---------------------------------------------------------------------------

typedef __attribute__((ext_vector_type(2))) float v2f;
typedef __attribute__((ext_vector_type(8))) float v8f;

#define NPOS   (512 * 2048)          // points per mode (positive half)
#define NFULL  (512 * 4096)          // full spectrum / signal points
#define TWO_PI 6.28318530717958647f

__device__ const float COS16[16] = {
     1.0f,  0.92387953251f,  0.70710678119f,  0.38268343236f,
     0.0f, -0.38268343236f, -0.70710678119f, -0.92387953251f,
    -1.0f, -0.92387953251f, -0.70710678119f, -0.38268343236f,
     0.0f,  0.38268343236f,  0.70710678119f,  0.92387953251f };
__device__ const float SIN16[16] = {
     0.0f,  0.38268343236f,  0.70710678119f,  0.92387953251f,
     1.0f,  0.92387953251f,  0.70710678119f,  0.38268343236f,
     0.0f, -0.38268343236f, -0.70710678119f, -0.92387953251f,
    -1.0f, -0.92387953251f, -0.70710678119f, -0.38268343236f };

// ---------------------------------------------------------------------------
// Build mirror-extended signal fM (512 x 4096), imag = 0.
//   fM[c][t]: t<1024 -> f[1023-t]; t<3072 -> f[t-1024]; else f[2047-(t-3072)]
//   f[c][t] = x[b][t][ch],  c = b*64 + ch
// ---------------------------------------------------------------------------
__global__ void build_fm_kernel(const float* __restrict__ x,
                                float2* __restrict__ buf) {
  int i = blockIdx.x * blockDim.x + threadIdx.x;   // [0, NFULL)
  int c = i >> 12, t = i & 4095;
  int ts = (t < 1024) ? (1023 - t) : (t < 3072 ? (t - 1024) : (2047 - (t - 3072)));
  int b = c >> 6, ch = c & 63;
  buf[i] = make_float2(x[b * 131072 + ts * 64 + ch], 0.0f);
}

// ---------------------------------------------------------------------------
// One radix-16 FFT stage over all 512 signals.
// One wave = one 16x16 complex tile: D = W16 * S, times stage twiddles.
// 16 v_wmma_f32_16x16x4_f32 per tile (complex matmul, K=16 in 4 chunks).
//   stage 1: contract a (n = a*256+b*16+cc), tile index t = b,  tw = w256^(b*k0)
//   stage 2: contract b,                      tile index t = k0, tw = w4096^(cc*(k0+16*k1))
//   stage 3: contract cc,                     tile index t = k0, no twiddle;
//            output bin = k0 + 16*k1 + 256*k2
// ---------------------------------------------------------------------------
__global__ void fft_stage_kernel(const float2* __restrict__ in,
                                 float2* __restrict__ out, int stage) {
  int wave = blockIdx.x * (blockDim.x >> 5) + (threadIdx.x >> 5); // [0,8192)
  int lane = threadIdx.x & 31;
  int c = wave >> 4;        // channel
  int t = wave & 15;        // b (stage1) or k0 (stage2/3)

  int base_in, sk, sn, base_out, orow, ocol;
  if (stage == 1)      { base_in = c*4096 + t*16;  sk = 256; sn = 1;
                         base_out = base_in;       orow = 256; ocol = 1; }
  else if (stage == 2) { base_in = c*4096 + t*256; sk = 16;  sn = 1;
                         base_out = base_in;       orow = 16;  ocol = 1; }
  else                 { base_in = c*4096 + t*256; sk = 1;   sn = 16;
                         base_out = c*4096 + t;    orow = 256; ocol = 16; }

  int m    = lane & 15;     // A row; also B/D column index n
  int h    = lane >> 4;
  int koff = 2 * h;         // K sub-index held by this lane (per ISA A layout)

  v8f dr = {}; v8f di = {};
#pragma unroll
  for (int kk = 0; kk < 4; ++kk) {
    int ka0 = 4 * kk + koff;
    int ka1 = ka0 + 1;
    // A = W16 chunk: W[m][k] = exp(-2*pi*i*m*k/16) = cos - i*sin
    float wr0 =  COS16[(m * ka0) & 15], wi0 = -SIN16[(m * ka0) & 15];
    float wr1 =  COS16[(m * ka1) & 15], wi1 = -SIN16[(m * ka1) & 15];
    v2f ar  = { wr0,  wr1};
    v2f ai  = { wi0,  wi1};
    v2f ani = {-wi0, -wi1};
    // B = data chunk: rows ka0/ka1 of contraction axis, column n = m
    float2 s0 = in[base_in + ka0 * sk + m * sn];
    float2 s1 = in[base_in + ka1 * sk + m * sn];
    v2f br = {s0.x, s1.x};
    v2f bi = {s0.y, s1.y};
    // Dr += Wr*Sr - Wi*Si ; Di += Wi*Sr + Wr*Si
    dr = __builtin_amdgcn_wmma_f32_16x16x4_f32(false, ar,  false, br, (short)0, dr, false, false);
    dr = __builtin_amdgcn_wmma_f32_16x16x4_f32(false, ani, false, bi, (short)0, dr, false, false);
    di = __builtin_amdgcn_wmma_f32_16x16x4_f32(false, ai,  false, br, (short)0, di, false, false);
    di = __builtin_amdgcn_wmma_f32_16x16x4_f32(false, ar,  false, bi, (short)0, di, false, false);
  }

  // Twiddle (e^{-i*phi}) and store. Lane holds rows j+8h at column n=m.
#pragma unroll
  for (int j = 0; j < 8; ++j) {
    int row = j + 8 * h;
    float re = dr[j], im = di[j];
    float ore = re, oim = im;
    if (stage != 3) {
      float phi = (stage == 1)
                    ? TWO_PI * (float)(t * row) * (1.0f / 256.0f)
                    : TWO_PI * (float)(m * (t + 16 * row)) * (1.0f / 4096.0f);
      float sp, cp;
      __sincosf(phi, &sp, &cp);
      ore = re * cp + im * sp;
      oim = im * cp - re * sp;
    }
    out[base_out + row * orow + m * ocol] = make_float2(ore, oim);
  }
}

// ---------------------------------------------------------------------------
__global__ void zero_u_kernel(float2* __restrict__ u) {
  int i = blockIdx.x * blockDim.x + threadIdx.x;   // [0, 4*NPOS)
  u[i] = make_float2(0.0f, 0.0f);
}

__global__ void init_sc_kernel(float* __restrict__ sc) {
  if (threadIdx.x == 0) {
    sc[0] = 0.0f; sc[1] = 0.125f; sc[2] = 0.25f; sc[3] = 0.375f;
    for (int i = 4; i < 14; ++i) sc[i] = 0.0f;     // num/den/diff/done
  }
}

// ---------------------------------------------------------------------------
// One MVMD iteration, fused across all 4 modes (pointwise in (c,f); only the
// omegas from the PREVIOUS iteration are consumed, matching the reference).
// fhp has row stride 4096 (aliased onto the forward-FFT output buffer).
// ---------------------------------------------------------------------------
__global__ void vmd_iter_kernel(const float2* __restrict__ fhp,
                                float2* __restrict__ u,
                                float* __restrict__ sc) {
  if (sc[13] != 0.0f) return;                      // converged: no-op (uniform)
  __shared__ float acc[9];                          // num[4], den[4], diff
  int tid = threadIdx.x;
  if (tid < 9) acc[tid] = 0.0f;
  __syncthreads();

  int idx = blockIdx.x * blockDim.x + tid;          // [0, NPOS), exact grid
  int c = idx >> 11;
  int f = idx & 2047;
  float fr = (float)f * (1.0f / 4096.0f);           // fpos[f]
  float2 F = fhp[c * 4096 + f];

  float2 uk[4];
  float  om[4];
#pragma unroll
  for (int k = 0; k < 4; ++k) { uk[k] = u[k * NPOS + idx]; om[k] = sc[k]; }

  float2 sum_all = make_float2(uk[0].x + uk[1].x + uk[2].x + uk[3].x,
                               uk[0].y + uk[1].y + uk[2].y + uk[3].y);
  float diff = 0.0f, num[4], den[4];
#pragma unroll
  for (int k = 0; k < 4; ++k) {
    float2 suk = make_float2(sum_all.x - uk[k].x, sum_all.y - uk[k].y);
    float d = fr - om[k];
    float inv = 1.0f / (1.0f + 2000.0f * d * d);    // ALPHA = 2000
    float2 nu = make_float2((F.x - suk.x) * inv, (F.y - suk.y) * inv);
    sum_all = make_float2(suk.x + nu.x, suk.y + nu.y);
    float dx = nu.x - uk[k].x, dy = nu.y - uk[k].y;
    diff += dx * dx + dy * dy;
    float pw = nu.x * nu.x + nu.y * nu.y;
    num[k] = fr * pw;
    den[k] = pw;
    u[k * NPOS + idx] = nu;
  }

  atomicAdd(&acc[8], diff);                          // ds_add_f32
#pragma unroll
  for (int k = 0; k < 4; ++k) {
    atomicAdd(&acc[k],     num[k]);
    atomicAdd(&acc[4 + k], den[k]);
  }
  __syncthreads();
  if (tid < 9) atomicAdd(&sc[4 + tid], acc[tid]);    // global_atomic_add_f32
}

__global__ void vmd_finalize_kernel(float* __restrict__ sc) {
  if (threadIdx.x != 0) return;
  if (sc[13] != 0.0f) return;
  for (int k = 0; k < 4; ++k) sc[k] = sc[4 + k] / sc[8 + k];  // omega update
  float uDiff = sc[12] * (1.0f / 4096.0f) + 1.1920929e-07f;
  if (uDiff <= 5e-5f) sc[13] = 1.0f;                 // TOL reached -> skip rest
  for (int i = 4; i < 13; ++i) sc[i] = 0.0f;         // reset accumulators
}

// ---------------------------------------------------------------------------
// Build conj(U) for mode 0 (Hermitian reconstruction folded with the conj of
// the inverse-FFT-via-forward-FFT trick):
//   U[j]    = u0[j]            j in [0,2048)
//   U[2048] = conj(u0[2047])
//   U[j]    = conj(u0[4096-j]) j in (2048,4096)
// store conj(U).
// ---------------------------------------------------------------------------
__global__ void build_spec_kernel(const float2* __restrict__ u0,
                                  float2* __restrict__ buf) {
  int i = blockIdx.x * blockDim.x + threadIdx.x;     // [0, NFULL)
  int c = i >> 12, j = i & 4095;
  float2 v;
  if (j < 2048)       { float2 a = u0[c * 2048 + j]; v = make_float2(a.x, -a.y); }
  else if (j == 2048) { v = u0[c * 2048 + 2047]; }
  else                { v = u0[c * 2048 + (4096 - j)]; }
  buf[i] = v;
}

// ---------------------------------------------------------------------------
// x_l[b,t,ch] = Re(FFT(conj(U))[c][1024+t]) / 4096 ;  x_h = x - x_l
// out = [x_h (1M floats) | x_l (1M floats)]
// ---------------------------------------------------------------------------
__global__ void output_kernel(const float* __restrict__ x,
                              const float2* __restrict__ spec,
                              float* __restrict__ out) {
  int i = blockIdx.x * blockDim.x + threadIdx.x;     // [0, 1048576)
  int b = i >> 17;
  int r = i & 131071;
  int t = r >> 6, ch = r & 63;
  int c = b * 64 + ch;
  float xl = spec[c * 4096 + 1024 + t].x * (1.0f / 4096.0f);
  out[1048576 + i] = xl;
  out[i] = x[i] - xl;
}

// ---------------------------------------------------------------------------
extern "C" void kernel_launch(void* const* d_in, const int* in_sizes, int n_in,
                              void* d_out, int out_size, void* d_ws, size_t ws_size,
                              hipStream_t stream) {
  const float* x = (const float*)d_in[0];
  float* out = (float*)d_out;

  char* ws = (char*)d_ws;
  float2* buf0 = (float2*)ws;                                   // 16MB
  float2* buf1 = (float2*)(ws + (size_t)NFULL * 8);             // 16MB
  float2* u    = (float2*)(ws + (size_t)2 * NFULL * 8);         // 32MB
  float*  sc   = (float*) (ws + (size_t)2 * NFULL * 8 + (size_t)4 * NPOS * 8);
  const float2* fhp = buf1;  // forward-FFT result, row stride 4096, bins 0..2047

  // Forward path: mirror-extend -> 3-stage radix-16 WMMA FFT
  build_fm_kernel<<<NFULL / 256, 256, 0, stream>>>(x, buf0);
  fft_stage_kernel<<<2048, 128, 0, stream>>>(buf0, buf1, 1);
  fft_stage_kernel<<<2048, 128, 0, stream>>>(buf1, buf0, 2);
  fft_stage_kernel<<<2048, 128, 0, stream>>>(buf0, buf1, 3);    // fhp = buf1

  // MVMD iterations (max 5, early-exit via done flag)
  zero_u_kernel<<<4 * NPOS / 256, 256, 0, stream>>>(u);
  init_sc_kernel<<<1, 32, 0, stream>>>(sc);
  for (int it = 0; it < 5; ++it) {
    vmd_iter_kernel<<<NPOS / 256, 256, 0, stream>>>(fhp, u, sc);
    vmd_finalize_kernel<<<1, 32, 0, stream>>>(sc);
  }

  // Inverse path for mode 0: conj-spectrum -> forward FFT -> Re()/N
  build_spec_kernel<<<NFULL / 256, 256, 0, stream>>>(u, buf0);
  fft_stage_kernel<<<2048, 128, 0, stream>>>(buf0, buf1, 1);    // clobbers fhp (done with it)
  fft_stage_kernel<<<2048, 128, 0, stream>>>(buf1, buf0, 2);
  fft_stage_kernel<<<2048, 128, 0, stream>>>(buf0, buf1, 3);

  output_kernel<<<1048576 / 256, 256, 0, stream>>>(x, buf1, out);
}